// GCNLayer_40398462386752
// MI455X (gfx1250) — compile-verified
//
#include <hip/hip_runtime.h>

typedef __attribute__((ext_vector_type(2))) float v2f;
typedef __attribute__((ext_vector_type(8))) float v8f;

#define F_DIM 128
#define TILE_M 64
#define XS_STRIDE 132   // pad so column reads (stride-132) spread across LDS banks

// ---------------------------------------------------------------------------
// Kernel 1: row_ptr[n] = lower_bound(edge_row, n)   (edge_row is sorted)
// ---------------------------------------------------------------------------
__global__ __launch_bounds__(256) void gcn_rowptr_kernel(
    const int* __restrict__ edge_row, int* __restrict__ row_ptr,
    int n_nodes, int n_edges) {
  int n = blockIdx.x * blockDim.x + threadIdx.x;
  if (n > n_nodes) return;
  int lo = 0, hi = n_edges;
  while (lo < hi) {
    int mid = (lo + hi) >> 1;
    if (edge_row[mid] < n) lo = mid + 1; else hi = mid;
  }
  row_ptr[n] = lo;
}

// ---------------------------------------------------------------------------
// Kernel 2: support = X @ W  via V_WMMA_F32_16X16X4_F32 (exact fp32)
// Block: 256 threads = 8 waves. Block tile: 64 rows x 128 cols.
// Wave w owns cols [16w, 16w+16) and 4 stacked 16x16 M-tiles.
// ---------------------------------------------------------------------------
__global__ __launch_bounds__(256) void gcn_gemm_wmma_kernel(
    const float* __restrict__ X, const float* __restrict__ W,
    float* __restrict__ S, int n_rows) {
  __shared__ float Xs[TILE_M * XS_STRIDE];

  const int t = threadIdx.x;
  const int row0 = blockIdx.x * TILE_M;
  const bool full_tile = (row0 + TILE_M) <= n_rows;   // block-uniform

  // Stage 64x128 X tile into LDS (float4).
  if (full_tile) {
#pragma unroll
    for (int i = 0; i < 8; ++i) {
      int idx = (t + i * 256) * 4;            // 0..8188, step 4
      int r = idx >> 7;                       // 0..63
      int c = idx & 127;
      float4 x = *(const float4*)(X + (size_t)(row0 + r) * F_DIM + c);
      *(float4*)(&Xs[r * XS_STRIDE + c]) = x;
    }
  } else {
#pragma unroll
    for (int i = 0; i < 8; ++i) {
      int idx = (t + i * 256) * 4;
      int r = idx >> 7;
      int c = idx & 127;
      int row = row0 + r;
      float4 x = make_float4(0.f, 0.f, 0.f, 0.f);
      if (row < n_rows) x = *(const float4*)(X + (size_t)row * F_DIM + c);
      *(float4*)(&Xs[r * XS_STRIDE + c]) = x;
    }
  }
  __syncthreads();

  const int wave = t >> 5;        // wave32 on gfx1250
  const int lane = t & 31;
  const int n0 = wave * 16;       // this wave's 16-column slice of W / output
  const int m  = lane & 15;       // A: M index; B/D: N index
  const int kh = lane >> 4;       // half-select: K {0,1} vs {2,3}; D: M vs M+8

  v8f acc[4] = {};
  for (int k = 0; k < F_DIM; k += 4) {
    // B fragment (4x16): VGPR0 = rows K=k+2kh, VGPR1 = rows K=k+2kh+1,
    // lanes 0-15 / 16-31 both carry N=0..15. Coalesced 64B reads; W is cached.
    v2f b;
    b.x = W[(k + 2 * kh)     * F_DIM + n0 + m];
    b.y = W[(k + 2 * kh + 1) * F_DIM + n0 + m];
#pragma unroll
    for (int mt = 0; mt < 4; ++mt) {
      // A fragment (16x4): lane m = row, kh selects K pair.
      const float* xp = &Xs[(mt * 16 + m) * XS_STRIDE + k + 2 * kh];
      v2f a; a.x = xp[0]; a.y = xp[1];
      acc[mt] = __builtin_amdgcn_wmma_f32_16x16x4_f32(
          false, a, false, b, (short)0, acc[mt], false, false);
    }
  }

  // D layout: VGPR i holds M = i (lanes 0-15) / M = i+8 (lanes 16-31), N = m.
  // One 64-bit base per wave; all 32 stores become constant-IOFFSET b32 stores.
  float* sp = S + (size_t)(row0 + 8 * kh) * F_DIM + (n0 + m);
  if (full_tile) {
#pragma unroll
    for (int mt = 0; mt < 4; ++mt) {
#pragma unroll
      for (int i = 0; i < 8; ++i) {
        sp[(mt * 16 + i) * F_DIM] = acc[mt][i];   // offset:K immediate
      }
    }
  } else {
#pragma unroll
    for (int mt = 0; mt < 4; ++mt) {
#pragma unroll
      for (int i = 0; i < 8; ++i) {
        int row = row0 + mt * 16 + i + 8 * kh;
        if (row < n_rows) sp[(mt * 16 + i) * F_DIM] = acc[mt][i];
      }
    }
  }
}

// ---------------------------------------------------------------------------
// Kernel 3: out[n] = bias + sum_{e in [row_ptr[n],row_ptr[n+1])} val[e]*support[col[e]]
// One wave32 per node; each lane owns a float4 slice of the 128 features.
// Deterministic (no atomics); support gathers hit L2 (51.2 MB < 192 MB).
// ---------------------------------------------------------------------------
__global__ __launch_bounds__(256) void gcn_spmm_kernel(
    const int* __restrict__ row_ptr, const int* __restrict__ edge_col,
    const float* __restrict__ edge_val, const float* __restrict__ support,
    const float* __restrict__ bias, float* __restrict__ out, int n_nodes) {
  int node = (int)((blockIdx.x * blockDim.x + threadIdx.x) >> 5);
  int lane = threadIdx.x & 31;
  if (node >= n_nodes) return;

  int e     = row_ptr[node];
  int e_end = row_ptr[node + 1];

  float4 acc = *((const float4*)bias + lane);
  for (; e < e_end; ++e) {
    int   c = edge_col[e];     // wave-uniform -> scalarized by compiler
    float v = edge_val[e];
    float4 s = *((const float4*)(support + (size_t)c * F_DIM) + lane);
    acc.x += v * s.x;
    acc.y += v * s.y;
    acc.z += v * s.z;
    acc.w += v * s.w;
  }
  *((float4*)(out + (size_t)node * F_DIM) + lane) = acc;
}

// ---------------------------------------------------------------------------
extern "C" void kernel_launch(void* const* d_in, const int* in_sizes, int n_in,
                              void* d_out, int out_size, void* d_ws, size_t ws_size,
                              hipStream_t stream) {
  (void)n_in; (void)out_size; (void)ws_size;

  const float* X        = (const float*)d_in[0];  // [N, 128]
  const int*   edge_row = (const int*)  d_in[1];  // [E] sorted
  const int*   edge_col = (const int*)  d_in[2];  // [E]
  const float* edge_val = (const float*)d_in[3];  // [E]
  const float* W        = (const float*)d_in[4];  // [128, 128]
  const float* bias     = (const float*)d_in[5];  // [128]

  const int N = in_sizes[0] / F_DIM;
  const int E = in_sizes[1];

  // Workspace layout: support [N*128 f32] | row_ptr [(N+1) i32]
  float* support = (float*)d_ws;
  int*   row_ptr = (int*)((char*)d_ws + (size_t)N * F_DIM * sizeof(float));
  float* out     = (float*)d_out;

  gcn_rowptr_kernel<<<(N + 256) / 256, 256, 0, stream>>>(edge_row, row_ptr, N, E);
  gcn_gemm_wmma_kernel<<<(N + TILE_M - 1) / TILE_M, 256, 0, stream>>>(X, W, support, N);
  gcn_spmm_kernel<<<(N + 7) / 8, 256, 0, stream>>>(row_ptr, edge_col, edge_val,
                                                   support, bias, out, N);
}